// ModifiedGAT_21157008900180
// MI455X (gfx1250) — compile-verified
//
#include <hip/hip_runtime.h>
#include <math.h>

// ---------------------------------------------------------------------------
// ModifiedGAT forward for MI455X (gfx1250, wave32).
//   - dense 128x128 GEMMs via V_WMMA_F32_16X16X4_F32 (exact fp32)
//   - edge phase: wave-per-edge float4 gathers (L2-resident), atomic softmax
//   - graph-layernorm via two-pass global reduction
// ---------------------------------------------------------------------------

typedef float v2f __attribute__((ext_vector_type(2)));
typedef float v8f __attribute__((ext_vector_type(8)));

#define NEG_ATT 0.2f
#define NEG_SLP 0.01f
#define LN_EPS  1e-5f

__device__ __forceinline__ float waveSum(float v) {
#pragma unroll
    for (int off = 16; off > 0; off >>= 1) v += __shfl_xor(v, off, 32);
    return v;
}

// float atomic-max via sign trick (init target to -inf)
__device__ __forceinline__ void atomicMaxF(float* a, float v) {
    if (v >= 0.f) atomicMax((int*)a, __float_as_int(v));
    else          atomicMin((unsigned int*)a, (unsigned int)__float_as_int(v));
}

// ---------------------------------------------------------------------------
// C[M,128] = A[M,128] @ W[128,128] + bias ; block = 16 rows x 128 cols,
// 8 waves per block, each wave owns one 16x16 tile; K loop in steps of 4.
// M must be a multiple of 16 (50000 = 3125*16). Tail handled by gemm_small.
// ---------------------------------------------------------------------------
__global__ __launch_bounds__(256) void gemm128_wmma(
    const float* __restrict__ A, const float* __restrict__ W,
    const float* __restrict__ bias, float* __restrict__ C)
{
    const int K = 128, Nc = 128;
    const int wave = threadIdx.x >> 5;
    const int lane = threadIdx.x & 31;
    const int row0 = blockIdx.x * 16;
    const int col0 = wave * 16;
    const int half = lane >> 4;   // 0 | 1
    const int l16  = lane & 15;
    const int kb   = half * 2;    // K sub-offset for this lane half

    const int m = row0 + l16;     // A row held by this lane
    const int n = col0 + l16;     // B/C column held by this lane

    v8f c;
    const float bv = bias[n];
#pragma unroll
    for (int r = 0; r < 8; ++r) c[r] = bv;

    const float* arow = A + (size_t)m * K;
#pragma unroll 4
    for (int k = 0; k < K; k += 4) {
        v2f a, b;
        a[0] = arow[k + kb];
        a[1] = arow[k + kb + 1];
        b[0] = W[(size_t)(k + kb)     * Nc + n];
        b[1] = W[(size_t)(k + kb + 1) * Nc + n];
        c = __builtin_amdgcn_wmma_f32_16x16x4_f32(
                false, a, false, b, (short)0, c, false, false);
    }
    // VGPR r holds (M = row0 + r + 8*half, N = n)
#pragma unroll
    for (int r = 0; r < 8; ++r)
        C[(size_t)(row0 + r + 8 * half) * Nc + n] = c[r];
}

// ---------------------------------------------------------------------------
// Generic small GEMM: C = [addend +] A@W + bias [, leaky]. One thread/output.
// ---------------------------------------------------------------------------
__global__ __launch_bounds__(256) void gemm_small(
    const float* __restrict__ A, const float* __restrict__ W,
    const float* __restrict__ bias, const float* __restrict__ addend,
    float* __restrict__ C, int M, int K, int Nc, int leaky)
{
    int idx = blockIdx.x * 256 + threadIdx.x;
    if (idx >= M * Nc) return;
    int m = idx / Nc, n = idx % Nc;
    float acc = bias[n];
    const float* a = A + (size_t)m * K;
    for (int k = 0; k < K; ++k) acc += a[k] * W[(size_t)k * Nc + n];
    if (addend) acc += addend[idx];
    if (leaky) acc = acc > 0.f ? acc : NEG_SLP * acc;
    C[idx] = acc;
}

__global__ __launch_bounds__(256) void zero_k(float* p, int n) {
    int i = blockIdx.x * 256 + threadIdx.x;
    if (i < n) p[i] = 0.f;
}

__global__ __launch_bounds__(256) void layer_init(
    float* outb, float* lmax, float* denom, float* stats, int N, int total)
{
    int i = blockIdx.x * 256 + threadIdx.x;
    if (i < total) outb[i] = 0.f;
    if (i < N) { lmax[i] = -__builtin_inff(); denom[i] = 0.f; }
    if (i < 2) stats[i] = 0.f;
}

// ---------------------------------------------------------------------------
// Wave per edge: logit_e = att . leaky(xl[src]+xr[dst], 0.2); atomic seg-max.
// ---------------------------------------------------------------------------
__global__ __launch_bounds__(256) void edge_logits_k(
    const float* __restrict__ xl, const float* __restrict__ xr,
    const int* __restrict__ esrc, const int* __restrict__ edst,
    const float* __restrict__ att, float* __restrict__ lg,
    float* __restrict__ lmax, int E, int EA)
{
    int e = (blockIdx.x * 256 + threadIdx.x) >> 5;   // uniform per wave
    int lane = threadIdx.x & 31;
    if (e >= EA) return;
    int s = (e < E) ? esrc[e] : (e - E);
    int d = (e < E) ? edst[e] : (e - E);
    float4 a = ((const float4*)(xl + (size_t)s * 128))[lane];
    float4 b = ((const float4*)(xr + (size_t)d * 128))[lane];
    float4 w = ((const float4*)att)[lane];
    float m0 = a.x + b.x; m0 = m0 > 0.f ? m0 : NEG_ATT * m0;
    float m1 = a.y + b.y; m1 = m1 > 0.f ? m1 : NEG_ATT * m1;
    float m2 = a.z + b.z; m2 = m2 > 0.f ? m2 : NEG_ATT * m2;
    float m3 = a.w + b.w; m3 = m3 > 0.f ? m3 : NEG_ATT * m3;
    float p = waveSum(m0 * w.x + m1 * w.y + m2 * w.z + m3 * w.w);
    if (lane == 0) { lg[e] = p; atomicMaxF(lmax + d, p); }
}

__global__ __launch_bounds__(256) void edge_exp_k(
    float* __restrict__ lg, const float* __restrict__ lmax,
    float* __restrict__ denom, const int* __restrict__ edst, int E, int EA)
{
    int e = blockIdx.x * 256 + threadIdx.x;
    if (e >= EA) return;
    int d = (e < E) ? edst[e] : (e - E);
    float ex = expf(lg[e] - lmax[d]);
    lg[e] = ex;
    atomicAdd(denom + d, ex);
}

__global__ __launch_bounds__(256) void edge_scatter_k(
    const float* __restrict__ xl, const float* __restrict__ lg,
    const float* __restrict__ denom, const int* __restrict__ esrc,
    const int* __restrict__ edst, float* __restrict__ outb, int E, int EA)
{
    int e = (blockIdx.x * 256 + threadIdx.x) >> 5;
    int lane = threadIdx.x & 31;
    if (e >= EA) return;
    int s = (e < E) ? esrc[e] : (e - E);
    int d = (e < E) ? edst[e] : (e - E);
    float alpha = lg[e] / denom[d];
    float4 v = ((const float4*)(xl + (size_t)s * 128))[lane];
    float* o = outb + (size_t)d * 128 + lane * 4;
    atomicAdd(o + 0, alpha * v.x);
    atomicAdd(o + 1, alpha * v.y);
    atomicAdd(o + 2, alpha * v.z);
    atomicAdd(o + 3, alpha * v.w);
}

// outb += conv_bias (per channel), leaky(0.01), accumulate sum / sum-of-sq.
__global__ __launch_bounds__(256) void post_stats_k(
    float* __restrict__ outb, const float* __restrict__ bias,
    float* __restrict__ stats, int total)
{
    __shared__ float ls[8], ls2[8];
    int idx = blockIdx.x * 256 + threadIdx.x;
    float t = 0.f;
    if (idx < total) {
        t = outb[idx] + bias[idx & 127];
        t = t > 0.f ? t : NEG_SLP * t;
        outb[idx] = t;
    }
    float s  = waveSum(t);
    float s2 = waveSum(t * t);
    int lane = threadIdx.x & 31, w = threadIdx.x >> 5;
    if (lane == 0) { ls[w] = s; ls2[w] = s2; }
    __syncthreads();
    if (w == 0) {
        float a = (lane < 8) ? ls[lane]  : 0.f;
        float b = (lane < 8) ? ls2[lane] : 0.f;
        a = waveSum(a); b = waveSum(b);
        if (lane == 0) { atomicAdd(&stats[0], a); atomicAdd(&stats[1], b); }
    }
}

// graph layernorm (whole-tensor mean/std, per-channel affine), D=128
__global__ __launch_bounds__(256) void ln_big_k(
    const float* __restrict__ outb, const float* __restrict__ stats,
    const float* __restrict__ g, const float* __restrict__ b,
    float* __restrict__ h, int total)
{
    int idx = blockIdx.x * 256 + threadIdx.x;
    if (idx >= total) return;
    float invM = 1.0f / (float)total;
    float mean = stats[0] * invM;
    float var  = stats[1] * invM - mean * mean;
    float sd   = sqrtf(fmaxf(var, 0.f)) + LN_EPS;
    h[idx] = (outb[idx] - mean) / sd * g[idx & 127] + b[idx & 127];
}

__global__ __launch_bounds__(256) void pool_k(
    const float* __restrict__ h, const int* __restrict__ batch,
    float* __restrict__ pooled, int total)
{
    int idx = blockIdx.x * 256 + threadIdx.x;
    if (idx >= total) return;
    atomicAdd(&pooled[(size_t)batch[idx >> 7] * 128 + (idx & 127)], h[idx]);
}

// single-block graph layernorm for small tensors
__global__ __launch_bounds__(256) void ln_small_k(
    float* __restrict__ x, const float* __restrict__ g,
    const float* __restrict__ b, int total, int D)
{
    __shared__ float ls[8], ls2[8], mb[2];
    float s = 0.f, s2 = 0.f;
    for (int i = threadIdx.x; i < total; i += 256) {
        float v = x[i]; s += v; s2 += v * v;
    }
    s = waveSum(s); s2 = waveSum(s2);
    int lane = threadIdx.x & 31, w = threadIdx.x >> 5;
    if (lane == 0) { ls[w] = s; ls2[w] = s2; }
    __syncthreads();
    if (threadIdx.x == 0) {
        float a = 0.f, c = 0.f;
        for (int i = 0; i < 8; ++i) { a += ls[i]; c += ls2[i]; }
        float mean = a / (float)total;
        float var  = c / (float)total - mean * mean;
        mb[0] = mean;
        mb[1] = sqrtf(fmaxf(var, 0.f)) + LN_EPS;
    }
    __syncthreads();
    float mean = mb[0], sd = mb[1];
    for (int i = threadIdx.x; i < total; i += 256)
        x[i] = (x[i] - mean) / sd * g[i % D] + b[i % D];
}

// ---------------------------------------------------------------------------
extern "C" void kernel_launch(void* const* d_in, const int* in_sizes, int n_in,
                              void* d_out, int out_size, void* d_ws, size_t ws_size,
                              hipStream_t stream)
{
    const float* x     = (const float*)d_in[0];
    const int*   ei    = (const int*)  d_in[1];
    const int*   batch = (const int*)  d_in[2];
    const float* gf    = (const float*)d_in[3];
    const float* cWl   = (const float*)d_in[4];
    const float* cbl   = (const float*)d_in[5];
    const float* cWr   = (const float*)d_in[6];
    const float* cbr   = (const float*)d_in[7];
    const float* catt  = (const float*)d_in[8];
    const float* cbias = (const float*)d_in[9];
    const float* cgam  = (const float*)d_in[10];
    const float* cbet  = (const float*)d_in[11];
    const float* mpW   = (const float*)d_in[12];
    const float* mpb   = (const float*)d_in[13];
    const float* mW    = (const float*)d_in[14];
    const float* mb    = (const float*)d_in[15];
    const float* mgam  = (const float*)d_in[16];
    const float* mbet  = (const float*)d_in[17];
    const float* pgW   = (const float*)d_in[18];
    const float* pgb   = (const float*)d_in[19];
    const float* f0W   = (const float*)d_in[20];
    const float* f0b   = (const float*)d_in[21];
    const float* fW    = (const float*)d_in[22];
    const float* fb    = (const float*)d_in[23];
    const float* oW    = (const float*)d_in[24];
    const float* ob    = (const float*)d_in[25];

    const int D  = 128;
    const int N  = in_sizes[0] / D;      // 50000
    const int E  = in_sizes[1] / 2;      // 800000
    const int B  = in_sizes[3] / D;      // 64
    const int EA = E + N;                // + self loops
    const int ND = N * D;

    float* ws = (float*)d_ws;
    size_t off = 0;
    auto take = [&](size_t n) { float* p = ws + off; off += n; return p; };
    float* hbuf   = take((size_t)ND);
    float* xl     = take((size_t)ND);
    float* xr     = take((size_t)ND);
    float* outb   = take((size_t)ND);
    float* lg     = take((size_t)EA);
    float* lmax   = take((size_t)N);
    float* denom  = take((size_t)N);
    float* stats  = take(8);
    float* pooled = take((size_t)B * D);
    float* g0     = take((size_t)B * 256);
    float* g1     = take((size_t)B * 256);
    float* a128   = take((size_t)B * D);
    float* s0     = take((size_t)B * 512);
    float* s1     = take((size_t)B * 512);
    (void)ws_size; (void)n_in; (void)out_size;

    const int* esrc = ei;
    const int* edst = ei + E;

    const int Mt = (N / 16) * 16;        // WMMA-covered rows (==N here)
    dim3 blk(256);
    const int gNd = (ND + 255) / 256;
    const int gEw = (EA + 7) / 8;        // wave-per-edge kernels
    const int gE  = (EA + 255) / 256;

    const float* hcur = x;
    for (int l = 0; l < 3; ++l) {
        const float* Wl = cWl + (size_t)l * D * D;
        const float* Wr = cWr + (size_t)l * D * D;
        gemm128_wmma<<<Mt / 16, blk, 0, stream>>>(hcur, Wl, cbl + l * D, xl);
        gemm128_wmma<<<Mt / 16, blk, 0, stream>>>(hcur, Wr, cbr + l * D, xr);
        if (Mt < N) {  // generic tail (not taken for N=50000)
            int Mr = N - Mt;
            gemm_small<<<(Mr * D + 255) / 256, blk, 0, stream>>>(
                hcur + (size_t)Mt * D, Wl, cbl + l * D, nullptr,
                xl + (size_t)Mt * D, Mr, D, D, 0);
            gemm_small<<<(Mr * D + 255) / 256, blk, 0, stream>>>(
                hcur + (size_t)Mt * D, Wr, cbr + l * D, nullptr,
                xr + (size_t)Mt * D, Mr, D, D, 0);
        }
        layer_init<<<gNd, blk, 0, stream>>>(outb, lmax, denom, stats, N, ND);
        edge_logits_k<<<gEw, blk, 0, stream>>>(xl, xr, esrc, edst,
                                               catt + l * D, lg, lmax, E, EA);
        edge_exp_k<<<gE, blk, 0, stream>>>(lg, lmax, denom, edst, E, EA);
        edge_scatter_k<<<gEw, blk, 0, stream>>>(xl, lg, denom, esrc, edst,
                                                outb, E, EA);
        post_stats_k<<<gNd, blk, 0, stream>>>(outb, cbias + l * D, stats, ND);
        ln_big_k<<<gNd, blk, 0, stream>>>(outb, stats, cgam + l * D,
                                          cbet + l * D, hbuf, ND);
        hcur = hbuf;
    }

    // sum-pool per graph
    zero_k<<<(B * D + 255) / 256, blk, 0, stream>>>(pooled, B * D);
    pool_k<<<gNd, blk, 0, stream>>>(hbuf, batch, pooled, ND);

    // global-feature MLP (64 rows — latency-trivial)
    gemm_small<<<(B * 256 + 255) / 256, blk, 0, stream>>>(gf, mpW, mpb, nullptr, g0, B, D, 256, 0);
    gemm_small<<<(B * 256 + 255) / 256, blk, 0, stream>>>(g0, mW, mb, nullptr, g1, B, 256, 256, 1);
    ln_small_k<<<1, blk, 0, stream>>>(g1, mgam, mbet, B * 256, 256);
    gemm_small<<<(B * 256 + 255) / 256, blk, 0, stream>>>(g1, mW + 256 * 256, mb + 256, nullptr, g0, B, 256, 256, 1);
    ln_small_k<<<1, blk, 0, stream>>>(g0, mgam + 256, mbet + 256, B * 256, 256);

    // a = pooled + g @ projg_W + projg_b ; then fc stack
    gemm_small<<<(B * D + 255) / 256, blk, 0, stream>>>(g0, pgW, pgb, pooled, a128, B, 256, D, 0);
    gemm_small<<<(B * 512 + 255) / 256, blk, 0, stream>>>(a128, f0W, f0b, nullptr, s0, B, D, 512, 1);
    gemm_small<<<(B * 512 + 255) / 256, blk, 0, stream>>>(s0, fW, fb, nullptr, s1, B, 512, 512, 1);
    gemm_small<<<(B * 512 + 255) / 256, blk, 0, stream>>>(s1, fW + 512 * 512, fb + 512, nullptr, s0, B, 512, 512, 1);
    gemm_small<<<1, blk, 0, stream>>>(s0, oW, ob, nullptr, (float*)d_out, B, 512, 1, 0);
}